// PoseOpt_83700322664775
// MI455X (gfx1250) — compile-verified
//
#include <hip/hip_runtime.h>
#include <hip/hip_bf16.h>
#include <math.h>

typedef float v2f __attribute__((ext_vector_type(2)));
typedef float v4f __attribute__((ext_vector_type(4)));
typedef float v8f __attribute__((ext_vector_type(8)));

#define NJ   24
#define DIN  88      // 72 rvec + 16 emb
#define HID  128
#define DOUT 75      // 72 rvec residual + 3 pelvis residual
#define RESS 0.1f

static constexpr int PAR[NJ] = {-1, 0, 0, 0, 1, 2, 3, 4, 5, 6, 7, 8,
                                 9, 9, 9, 12, 13, 14, 16, 17, 18, 19, 20, 21};

// =====================================================================
// Kernel 1: fused 4-layer MLP using V_WMMA_F32_16X16X4_F32 (exact fp32).
// One wave (32 lanes) owns a 16-row tile; hidden activations live in a
// per-wave 16x128 LDS buffer between layers.
// =====================================================================
__global__ __launch_bounds__(128) void poseopt_mlp(
    const float* __restrict__ bones,   // N x 72
    const float* __restrict__ emb,     // NU x 16
    const float* __restrict__ pelvis,  // NU x 3
    const float* __restrict__ W0, const float* __restrict__ b0,
    const float* __restrict__ W1, const float* __restrict__ b1,
    const float* __restrict__ W2, const float* __restrict__ b2,
    const float* __restrict__ W3, const float* __restrict__ b3,
    float* __restrict__ out_rvecs,     // NU x 72  (rvec + residual)
    float* __restrict__ out_pelvis,    // NU x 3   (pelvis + residual)
    int NU, int skip)
{
    __shared__ float Hs[4][16][HID];
    const int wave = threadIdx.x >> 5;
    const int lane = threadIdx.x & 31;
    const int half = lane >> 4;     // 0: K=k,k+1 side, 1: K=k+2,k+3 side
    const int m    = lane & 15;
    const int row0 = (blockIdx.x * 4 + wave) * 16;
    float (*H)[HID] = Hs[wave];

    v8f acc[8];

    // ---------------- layer 0: X(16x88) @ W0(88x128), +b0, relu ----------------
    #pragma unroll
    for (int nt = 0; nt < 8; ++nt)
        #pragma unroll
        for (int r = 0; r < 8; ++r) acc[nt][r] = 0.f;

    {
        int rA = row0 + m; if (rA > NU - 1) rA = NU - 1;
        const size_t boneRow = (size_t)rA * skip * (NJ * 3);
        #pragma unroll 2
        for (int k = 0; k < DIN; k += 4) {
            const int c0 = k + 2 * half;
            v2f a;
            a.x = (c0     < NJ*3) ? bones[boneRow + c0]     : emb[(size_t)rA*16 + (c0     - NJ*3)];
            a.y = (c0 + 1 < NJ*3) ? bones[boneRow + c0 + 1] : emb[(size_t)rA*16 + (c0 + 1 - NJ*3)];
            #pragma unroll
            for (int nt = 0; nt < 8; ++nt) {
                v2f b;
                b.x = W0[(size_t)c0       * HID + nt*16 + m];
                b.y = W0[(size_t)(c0 + 1) * HID + nt*16 + m];
                acc[nt] = __builtin_amdgcn_wmma_f32_16x16x4_f32(
                    false, a, false, b, (short)0, acc[nt], false, false);
            }
        }
    }
    #pragma unroll
    for (int nt = 0; nt < 8; ++nt)
        #pragma unroll
        for (int r = 0; r < 8; ++r) {
            const int col = nt*16 + m;
            H[r + 8*half][col] = fmaxf(acc[nt][r] + b0[col], 0.f);
        }
    __syncthreads();

    // ---------------- layers 1 & 2: H(16x128) @ W(128x128), +b, relu ----------
    for (int layer = 0; layer < 2; ++layer) {
        const float* Wl = layer ? W2 : W1;
        const float* bl = layer ? b2 : b1;
        #pragma unroll
        for (int nt = 0; nt < 8; ++nt)
            #pragma unroll
            for (int r = 0; r < 8; ++r) acc[nt][r] = 0.f;
        #pragma unroll 4
        for (int k = 0; k < HID; k += 4) {
            const int c0 = k + 2 * half;
            v2f a = *(const v2f*)&H[m][c0];
            #pragma unroll
            for (int nt = 0; nt < 8; ++nt) {
                v2f b;
                b.x = Wl[(size_t)c0       * HID + nt*16 + m];
                b.y = Wl[(size_t)(c0 + 1) * HID + nt*16 + m];
                acc[nt] = __builtin_amdgcn_wmma_f32_16x16x4_f32(
                    false, a, false, b, (short)0, acc[nt], false, false);
            }
        }
        __syncthreads();
        #pragma unroll
        for (int nt = 0; nt < 8; ++nt)
            #pragma unroll
            for (int r = 0; r < 8; ++r) {
                const int col = nt*16 + m;
                H[r + 8*half][col] = fmaxf(acc[nt][r] + bl[col], 0.f);
            }
        __syncthreads();
    }

    // ---------------- layer 3: H(16x128) @ W3(128x75), +b3, *0.1 --------------
    v8f acc3[5];
    #pragma unroll
    for (int nt = 0; nt < 5; ++nt)
        #pragma unroll
        for (int r = 0; r < 8; ++r) acc3[nt][r] = 0.f;
    #pragma unroll 4
    for (int k = 0; k < HID; k += 4) {
        const int c0 = k + 2 * half;
        v2f a = *(const v2f*)&H[m][c0];
        #pragma unroll
        for (int nt = 0; nt < 5; ++nt) {
            const int col = nt*16 + m;
            v2f b;
            b.x = (col < DOUT) ? W3[(size_t)c0       * DOUT + col] : 0.f;
            b.y = (col < DOUT) ? W3[(size_t)(c0 + 1) * DOUT + col] : 0.f;
            acc3[nt] = __builtin_amdgcn_wmma_f32_16x16x4_f32(
                false, a, false, b, (short)0, acc3[nt], false, false);
        }
    }
    #pragma unroll
    for (int nt = 0; nt < 5; ++nt)
        #pragma unroll
        for (int r = 0; r < 8; ++r) {
            const int col = nt*16 + m;
            const int row = row0 + r + 8*half;
            if (col < DOUT && row < NU) {
                const float res = (acc3[nt][r] + b3[col]) * RESS;
                if (col < NJ*3)
                    out_rvecs[(size_t)row*(NJ*3) + col] =
                        bones[(size_t)row*skip*(NJ*3) + col] + res;
                else
                    out_pelvis[(size_t)row*3 + (col - NJ*3)] =
                        pelvis[(size_t)row*3 + (col - NJ*3)] + res;
            }
        }
}

// =====================================================================
// Kernel 2: forward kinematics, one thread per unique sample.
// Joint loop fully unrolled with constexpr parents -> parent transforms
// stay in registers (max live set ~8 matrices).
// =====================================================================
__global__ __launch_bounds__(256) void poseopt_kin(
    const float* __restrict__ rvecs,     // NU x 72 (post-residual)
    const float* __restrict__ pelvis,    // NU x 3  (post-residual)
    const float* __restrict__ rest_pose, // 24 x 3
    float* __restrict__ kp_u,            // NU x 24 x 3
    float* __restrict__ skts_u,          // NU x 24 x 16
    int NU)
{
    const int u = blockIdx.x * blockDim.x + threadIdx.x;
    if (u >= NU) return;
    const float* rv = rvecs + (size_t)u * (NJ*3);

    float Rw[NJ][9];
    float tw[NJ][3];

    #pragma unroll
    for (int j = 0; j < NJ; ++j) {
        // ---- Rodrigues ----
        const float x0 = rv[j*3+0], y0 = rv[j*3+1], z0 = rv[j*3+2];
        const float th  = sqrtf(x0*x0 + y0*y0 + z0*z0);
        const float inv = 1.0f / fmaxf(th, 1e-8f);
        const float ax = x0*inv, ay = y0*inv, az = z0*inv;
        const float s = sinf(th), c = cosf(th), t1 = 1.0f - c;
        float R[9];
        R[0] = c + t1*ax*ax;      R[1] = t1*ax*ay - s*az;  R[2] = t1*ax*az + s*ay;
        R[3] = t1*ax*ay + s*az;   R[4] = c + t1*ay*ay;     R[5] = t1*ay*az - s*ax;
        R[6] = t1*ax*az - s*ay;   R[7] = t1*ay*az + s*ax;  R[8] = c + t1*az*az;

        if (j == 0) {
            #pragma unroll
            for (int i = 0; i < 9; ++i) Rw[0][i] = R[i];
            tw[0][0] = pelvis[(size_t)u*3+0];
            tw[0][1] = pelvis[(size_t)u*3+1];
            tw[0][2] = pelvis[(size_t)u*3+2];
        } else {
            const int p = PAR[j];
            const float tl0 = rest_pose[j*3+0] - rest_pose[p*3+0];
            const float tl1 = rest_pose[j*3+1] - rest_pose[p*3+1];
            const float tl2 = rest_pose[j*3+2] - rest_pose[p*3+2];
            #pragma unroll
            for (int rr = 0; rr < 3; ++rr) {
                #pragma unroll
                for (int cc = 0; cc < 3; ++cc)
                    Rw[j][rr*3+cc] = Rw[p][rr*3+0]*R[0+cc]
                                   + Rw[p][rr*3+1]*R[3+cc]
                                   + Rw[p][rr*3+2]*R[6+cc];
                tw[j][rr] = Rw[p][rr*3+0]*tl0 + Rw[p][rr*3+1]*tl1
                          + Rw[p][rr*3+2]*tl2 + tw[p][rr];
            }
        }

        // ---- kp + inverse transform (skts) ----
        const float kx = tw[j][0], ky = tw[j][1], kz = tw[j][2];
        float* kp = kp_u + ((size_t)u*NJ + j)*3;
        kp[0] = kx; kp[1] = ky; kp[2] = kz;

        float* S = skts_u + ((size_t)u*NJ + j)*16;
        S[0]  = Rw[j][0]; S[1]  = Rw[j][3]; S[2]  = Rw[j][6];
        S[3]  = -(Rw[j][0]*kx + Rw[j][3]*ky + Rw[j][6]*kz);
        S[4]  = Rw[j][1]; S[5]  = Rw[j][4]; S[6]  = Rw[j][7];
        S[7]  = -(Rw[j][1]*kx + Rw[j][4]*ky + Rw[j][7]*kz);
        S[8]  = Rw[j][2]; S[9]  = Rw[j][5]; S[10] = Rw[j][8];
        S[11] = -(Rw[j][2]*kx + Rw[j][5]*ky + Rw[j][8]*kz);
        S[12] = 0.f; S[13] = 0.f; S[14] = 0.f; S[15] = 1.f;
    }
}

// =====================================================================
// Kernel 3: replication (jnp.repeat x skip), one templated instance per
// output region. One thread per SOURCE 16B vector -> reads once, writes
// skip replicas with non-temporal stores (streaming 277MB must not
// thrash the 192MB L2). 32-bit indexing, constant-divisor div/mod.
// Uses native ext_vector v4f (HIP float4 class is rejected by
// __builtin_nontemporal_store).
// =====================================================================
template <int ROW4>
__global__ __launch_bounds__(256) void poseopt_rep(
    const v4f* __restrict__ src,   // NU * ROW4 v4f
    v4f* __restrict__ dst,         // N  * ROW4 v4f
    int NU, int skip)
{
    const int i = blockIdx.x * blockDim.x + threadIdx.x;
    if (i >= NU * ROW4) return;
    const int u = i / ROW4;           // constant divisor -> mul/shift
    const int r = i - u * ROW4;
    const v4f v = src[i];
    const int base = u * skip * ROW4 + r;
    #pragma unroll
    for (int rep = 0; rep < 8; ++rep) {           // skip == 8 in practice
        if (rep < skip)
            __builtin_nontemporal_store(v, &dst[base + rep * ROW4]);
    }
    // generic tail if skip > 8
    for (int rep = 8; rep < skip; ++rep)
        __builtin_nontemporal_store(v, &dst[base + rep * ROW4]);
}

extern "C" void kernel_launch(void* const* d_in, const int* in_sizes, int n_in,
                              void* d_out, int out_size, void* d_ws, size_t ws_size,
                              hipStream_t stream)
{
    (void)n_in; (void)out_size; (void)ws_size;
    const float* bones  = (const float*)d_in[0];
    // d_in[1] kp3d, d_in[2] kp_idxs: not needed (kp_idxs[::skip] == arange(NU))
    const float* emb    = (const float*)d_in[3];
    const float* pelvis = (const float*)d_in[4];
    const float* rest   = (const float*)d_in[5];
    const float* W0 = (const float*)d_in[6];  const float* b0 = (const float*)d_in[7];
    const float* W1 = (const float*)d_in[8];  const float* b1 = (const float*)d_in[9];
    const float* W2 = (const float*)d_in[10]; const float* b2 = (const float*)d_in[11];
    const float* W3 = (const float*)d_in[12]; const float* b3 = (const float*)d_in[13];

    const int N    = in_sizes[0] / (NJ * 3);   // bones: N x 24 x 3
    const int NU   = in_sizes[4] / 3;          // pelvis_buf: NU x 3
    const int skip = N / NU;

    // workspace partition (floats): rvecs | pelvis | kp_u | skts_u
    float* ws     = (float*)d_ws;
    float* ws_rv  = ws;                         // NU*72
    float* ws_pel = ws + (size_t)NU * 72;       // NU*3
    float* ws_kp  = ws + (size_t)NU * 75;       // NU*72
    float* ws_skt = ws + (size_t)NU * 147;      // NU*384

    const int tiles     = (NU + 15) / 16;
    const int mlpBlocks = (tiles + 3) / 4;
    poseopt_mlp<<<mlpBlocks, 128, 0, stream>>>(
        bones, emb, pelvis, W0, b0, W1, b1, W2, b2, W3, b3,
        ws_rv, ws_pel, NU, skip);

    poseopt_kin<<<(NU + 255) / 256, 256, 0, stream>>>(
        ws_rv, ws_pel, rest, ws_kp, ws_skt, NU);

    // d_out layout (v4f units): kp N*18 | skts N*96 | rvecs N*18
    v4f* out4 = (v4f*)d_out;
    poseopt_rep<18><<<(NU * 18 + 255) / 256, 256, 0, stream>>>(
        (const v4f*)ws_kp, out4, NU, skip);
    poseopt_rep<96><<<(NU * 96 + 255) / 256, 256, 0, stream>>>(
        (const v4f*)ws_skt, out4 + (size_t)N * 18, NU, skip);
    poseopt_rep<18><<<(NU * 18 + 255) / 256, 256, 0, stream>>>(
        (const v4f*)ws_rv, out4 + (size_t)N * 114, NU, skip);
}